// TransitionDown_3375844295199
// MI455X (gfx1250) — compile-verified
//
#include <hip/hip_runtime.h>

#define N_PTS   16384
#define C_IN    64
#define C_OUT   128
#define KNN_K   16
#define M_CL    4096
#define FAN     67          // 3 + C_IN
#define FAN_PAD 68          // padded to multiple of 4 for wmma K=4
#define EDGES   (M_CL * KNN_K)   // 65536
#define BN_EPS  1e-5f

typedef __attribute__((ext_vector_type(2))) float v2f;
typedef __attribute__((ext_vector_type(8))) float v8f;
typedef unsigned long long u64;

// ---------------------------------------------------------------------------
// 1) Farthest point sampling: single workgroup; pos AND min_d fully LDS-
//    resident (256.3 KB of the 320 KB WGP LDS); two-level shuffle argmax
//    (2 barriers/iter instead of 11).
// ---------------------------------------------------------------------------
__global__ __launch_bounds__(1024) void fps_kernel(const float* __restrict__ pos,
                                                   int* __restrict__ clusters) {
    __shared__ float px[N_PTS];       // 64 KB
    __shared__ float py[N_PTS];       // 64 KB
    __shared__ float pz[N_PTS];       // 64 KB
    __shared__ float mind[N_PTS];     // 64 KB
    __shared__ float wv[32];
    __shared__ int   wi[32];
    __shared__ int   last_sh;
    const int tid  = threadIdx.x;
    const int lane = tid & 31;
    const int wav  = tid >> 5;

    for (int i = tid; i < N_PTS; i += 1024) {
        px[i] = pos[i * 3 + 0];
        py[i] = pos[i * 3 + 1];
        pz[i] = pos[i * 3 + 2];
        mind[i] = __builtin_inff();
    }
    if (tid == 0) { clusters[0] = 0; last_sh = 0; }
    __syncthreads();

    int last = 0;
    for (int it = 1; it < M_CL; ++it) {
        const float lx = px[last];
        const float ly = py[last];
        const float lz = pz[last];
        float best = -1.0f; int bidx = 0;
#pragma unroll
        for (int r = 0; r < N_PTS / 1024; ++r) {
            const int i = tid + r * 1024;
            float dx = px[i] - lx;
            float dy = py[i] - ly;
            float dz = pz[i] - lz;
            float d  = dx * dx + dy * dy + dz * dz;
            float m  = fminf(mind[i], d);
            mind[i]  = m;
            if (m > best) { best = m; bidx = i; }   // strict > keeps first max
        }
        // wave32 butterfly argmax (max value, tie -> min index)
        for (int off = 16; off >= 1; off >>= 1) {
            float ov = __shfl_xor(best, off, 32);
            int   oi = __shfl_xor(bidx, off, 32);
            if (ov > best || (ov == best && oi < bidx)) { best = ov; bidx = oi; }
        }
        if (lane == 0) { wv[wav] = best; wi[wav] = bidx; }
        __syncthreads();
        if (wav == 0) {
            float v  = wv[lane];
            int   bi = wi[lane];
            for (int off = 16; off >= 1; off >>= 1) {
                float ov = __shfl_xor(v, off, 32);
                int   oi = __shfl_xor(bi, off, 32);
                if (ov > v || (ov == v && oi < bi)) { v = ov; bi = oi; }
            }
            if (lane == 0) { last_sh = bi; clusters[it] = bi; }
        }
        __syncthreads();
        last = last_sh;
    }
}

// ---------------------------------------------------------------------------
// 2) kNN: one wave32 per query; packed (dist_bits<<32 | idx) keys give
//    stable smallest-first selection matching lax.top_k tie-breaking.
// ---------------------------------------------------------------------------
__device__ inline u64 shfl_xor_u64(u64 v, int mask) {
    int lo = (int)(unsigned)(v & 0xffffffffull);
    int hi = (int)(unsigned)(v >> 32);
    lo = __shfl_xor(lo, mask, 32);
    hi = __shfl_xor(hi, mask, 32);
    return ((u64)(unsigned)hi << 32) | (u64)(unsigned)lo;
}

__global__ __launch_bounds__(256) void knn_kernel(const float* __restrict__ pos,
                                                  const int* __restrict__ clusters,
                                                  int* __restrict__ nbr) {
    const int wave = (int)((blockIdx.x * blockDim.x + threadIdx.x) >> 5);
    const int lane = threadIdx.x & 31;
    if (wave >= M_CL) return;

    const int   q  = clusters[wave];
    const float qx = pos[q * 3 + 0];
    const float qy = pos[q * 3 + 1];
    const float qz = pos[q * 3 + 2];

    u64 arr[KNN_K];
#pragma unroll
    for (int j = 0; j < KNN_K; ++j) arr[j] = ~0ull;

    for (int i = lane; i < N_PTS; i += 32) {
        float dx = pos[i * 3 + 0] - qx;
        float dy = pos[i * 3 + 1] - qy;
        float dz = pos[i * 3 + 2] - qz;
        float d  = dx * dx + dy * dy + dz * dz;
        u64 key  = ((u64)__float_as_uint(d) << 32) | (u64)(unsigned)i;
        if (key < arr[KNN_K - 1]) {
#pragma unroll
            for (int j = KNN_K - 1; j >= 1; --j) {
                u64 prev = arr[j - 1];
                arr[j] = (key < prev) ? prev : ((key < arr[j]) ? key : arr[j]);
            }
            arr[0] = (key < arr[0]) ? key : arr[0];
        }
    }

    // merge 32 sorted lists: 16 rounds of wave-min + pop winner's head
    for (int r = 0; r < KNN_K; ++r) {
        u64 best = arr[0];
        for (int off = 16; off >= 1; off >>= 1) {
            u64 other = shfl_xor_u64(best, off);
            if (other < best) best = other;
        }
        if (lane == 0) nbr[wave * KNN_K + r] = (int)(unsigned)(best & 0xffffffffull);
        const bool winner = (arr[0] == best);
#pragma unroll
        for (int j = 0; j < KNN_K - 1; ++j) arr[j] = winner ? arr[j + 1] : arr[j];
        arr[KNN_K - 1] = winner ? ~0ull : arr[KNN_K - 1];
    }
}

// ---------------------------------------------------------------------------
// 3) Gather + Linear(67->128) via V_WMMA_F32_16X16X4_F32, one wave per
//    16x16 output tile. h stored to workspace for BN stats / segment_max.
//    NOTE: faithful to reference, relative_pos = pos[n] - pos[query_id]
//    (query id indexes the FULL pos array, not pos[clusters[...]]).
// ---------------------------------------------------------------------------
__device__ inline float featval(float r0, float r1, float r2,
                                const float* __restrict__ xr, int k) {
    if (k >= FAN) return 0.0f;
    if (k >= 3)   return xr[k - 3];
    return (k == 0) ? r0 : ((k == 1) ? r1 : r2);
}

__global__ __launch_bounds__(256) void gemm_kernel(const float* __restrict__ x,
                                                   const float* __restrict__ pos,
                                                   const float* __restrict__ W,
                                                   const int* __restrict__ nbr,
                                                   float* __restrict__ h) {
    const int wid  = (int)((blockIdx.x * blockDim.x + threadIdx.x) >> 5); // 0..32767
    const int lane = threadIdx.x & 31;
    const int rt   = wid >> 3;     // 0..4095 row tiles
    const int ct   = wid & 7;      // 0..7   col tiles
    const int half = lane >> 4;    // 0 | 1
    const int l16  = lane & 15;

    // A-matrix row for this lane (edge index)
    const int row = rt * 16 + l16;
    const int n   = nbr[row];
    const int qi  = row >> 4;      // raw query id (reference quirk)
    const float r0 = pos[n * 3 + 0] - pos[qi * 3 + 0];
    const float r1 = pos[n * 3 + 1] - pos[qi * 3 + 1];
    const float r2 = pos[n * 3 + 2] - pos[qi * 3 + 2];
    const float* xr = x + (size_t)n * C_IN;

    // B-matrix column for this lane (output channel)
    const int bcol = ct * 16 + l16;
    const float* wr = W + (size_t)bcol * FAN;

    v8f acc = {0.f, 0.f, 0.f, 0.f, 0.f, 0.f, 0.f, 0.f};
    for (int kk = 0; kk < FAN_PAD; kk += 4) {
        const int k0 = kk + 2 * half;
        v2f av, bv;
        av.x = featval(r0, r1, r2, xr, k0);
        av.y = featval(r0, r1, r2, xr, k0 + 1);
        bv.x = (k0     < FAN) ? wr[k0]     : 0.0f;
        bv.y = (k0 + 1 < FAN) ? wr[k0 + 1] : 0.0f;
        acc = __builtin_amdgcn_wmma_f32_16x16x4_f32(
            /*neg_a=*/false, av, /*neg_b=*/false, bv,
            /*c_mod=*/(short)0, acc, /*reuse_a=*/false, /*reuse_b=*/false);
    }

    // C/D layout: VGPR j -> M = j + 8*half, N = l16
#pragma unroll
    for (int j = 0; j < 8; ++j) {
        const int orow = rt * 16 + j + 8 * half;
        h[(size_t)orow * C_OUT + bcol] = acc[j];
    }
}

// ---------------------------------------------------------------------------
// 4) BatchNorm training stats, deterministic per-channel tree reduction.
// ---------------------------------------------------------------------------
__global__ __launch_bounds__(256) void bnstats_kernel(const float* __restrict__ h,
                                                      float* __restrict__ mean,
                                                      float* __restrict__ rstd) {
    const int c   = blockIdx.x;    // 0..127
    const int tid = threadIdx.x;
    float s = 0.f, sq = 0.f;
    for (int r = tid; r < EDGES; r += 256) {
        float v = h[(size_t)r * C_OUT + c];
        s  += v;
        sq += v * v;
    }
    __shared__ float sh_s[256], sh_q[256];
    sh_s[tid] = s; sh_q[tid] = sq;
    __syncthreads();
    for (int st = 128; st > 0; st >>= 1) {
        if (tid < st) { sh_s[tid] += sh_s[tid + st]; sh_q[tid] += sh_q[tid + st]; }
        __syncthreads();
    }
    if (tid == 0) {
        float m   = sh_s[0] / (float)EDGES;
        float var = sh_q[0] / (float)EDGES - m * m;
        var = fmaxf(var, 0.0f);
        mean[c] = m;
        rstd[c] = rsqrtf(var + BN_EPS);
    }
}

// ---------------------------------------------------------------------------
// 5) Normalize + ReLU + segment_max(K=16) + sub_pos/sub_batch emission.
// ---------------------------------------------------------------------------
__global__ __launch_bounds__(128) void finish_kernel(const float* __restrict__ h,
                                                     const float* __restrict__ pos,
                                                     const int* __restrict__ clusters,
                                                     const float* __restrict__ gamma,
                                                     const float* __restrict__ beta,
                                                     const float* __restrict__ mean,
                                                     const float* __restrict__ rstd,
                                                     float* __restrict__ out) {
    const int m = blockIdx.x;      // 0..4095
    const int c = threadIdx.x;     // 0..127
    const float g  = gamma[c];
    const float b  = beta[c];
    const float mu = mean[c];
    const float rs = rstd[c];
    float mx = -__builtin_inff();
#pragma unroll
    for (int j = 0; j < KNN_K; ++j) {
        float v = h[(size_t)(m * KNN_K + j) * C_OUT + c];
        v = g * (v - mu) * rs + b;
        v = fmaxf(v, 0.0f);
        mx = fmaxf(mx, v);
    }
    out[(size_t)m * C_OUT + c] = mx;

    float* sub_pos   = out + (size_t)M_CL * C_OUT;
    float* sub_batch = sub_pos + (size_t)M_CL * 3;
    if (c < 3)  sub_pos[m * 3 + c] = pos[clusters[m] * 3 + c];
    if (c == 3) sub_batch[m] = 0.0f;   // int32 0 bit pattern
}

// ---------------------------------------------------------------------------
// Host launcher. Inputs: x, pos, batch, W, gamma, beta.
// d_out = [out 4096x128 | sub_pos 4096x3 | sub_batch 4096] flat.
// ---------------------------------------------------------------------------
extern "C" void kernel_launch(void* const* d_in, const int* in_sizes, int n_in,
                              void* d_out, int out_size, void* d_ws, size_t ws_size,
                              hipStream_t stream) {
    (void)in_sizes; (void)n_in; (void)out_size; (void)ws_size;
    const float* x     = (const float*)d_in[0];
    const float* pos   = (const float*)d_in[1];
    /* batch = d_in[2]: all zeros, unused */
    const float* W     = (const float*)d_in[3];
    const float* gamma = (const float*)d_in[4];
    const float* beta  = (const float*)d_in[5];

    char* ws = (char*)d_ws;
    int*   clusters = (int*)ws;                              // 16 KB
    int*   nbr      = (int*)(ws + 16384);                    // 256 KB
    float* h        = (float*)(ws + 16384 + 262144);         // 33.5 MB
    float* mean     = (float*)(ws + 16384 + 262144 + (size_t)EDGES * C_OUT * 4);
    float* rstd     = mean + C_OUT;
    float* outp     = (float*)d_out;

    fps_kernel    <<<1,        1024, 0, stream>>>(pos, clusters);
    knn_kernel    <<<M_CL / 8, 256,  0, stream>>>(pos, clusters, nbr);
    gemm_kernel   <<<M_CL,     256,  0, stream>>>(x, pos, W, nbr, h);  // 4096 blocks x 8 waves
    bnstats_kernel<<<C_OUT,    256,  0, stream>>>(h, mean, rstd);
    finish_kernel <<<M_CL,     128,  0, stream>>>(h, pos, clusters, gamma, beta, mean, rstd, outp);
}